// CBAM_43465069036200
// MI455X (gfx1250) — compile-verified
//
#include <hip/hip_runtime.h>
#include <math.h>

// Problem constants
#define B_  32
#define L_  4096
#define C_  256
#define H_  16          // C/R
#define LT_ 256         // l-rows per reduction block
#define NLB (L_ / LT_)  // 16 reduction blocks per batch
#define RPW (LT_ / 8)   // rows per wave (8 waves / block) = 32

typedef __attribute__((ext_vector_type(2))) float v2f;
typedef __attribute__((ext_vector_type(4))) float v4f;
typedef __attribute__((ext_vector_type(8))) float v8f;

// ---------------------------------------------------------------------------
// Kernel 1: fused reductions.  One block = one (b, l-tile) of [LT_ x C_].
// Wave-private spatial reduction: wave w owns rows w*32..w*32+31; each lane
// loads 8 contiguous channels (2x global_load_b128, full 1KB row per wave).
// Row mean/max finish with shfl only (zero barriers in the row loop).
// Channel partials cross waves via LDS exactly once per block.
// Side effect we want: this pass pulls all of x (128MB) into the 192MB L2.
// ---------------------------------------------------------------------------
__global__ __launch_bounds__(256) void reduce_kernel(
    const float* __restrict__ x,
    float* __restrict__ psum,   // [B_, NLB, C_]
    float* __restrict__ pmax,   // [B_, NLB, C_]
    float* __restrict__ avg_s,  // [B_, L_]
    float* __restrict__ max_s)  // [B_, L_]
{
    __shared__ float sSum[8][C_];   // 8KB
    __shared__ float sMax[8][C_];   // 8KB

    const int b    = blockIdx.x / NLB;
    const int lb   = blockIdx.x % NLB;
    const int lane = threadIdx.x & 31;
    const int wv   = threadIdx.x >> 5;

    const int l0 = lb * LT_ + wv * RPW;              // first row of this wave
    const float* xw = x + ((size_t)b * L_ + l0) * C_;

    float cs[8], cm[8];
    #pragma unroll
    for (int j = 0; j < 8; ++j) { cs[j] = 0.0f; cm[j] = -3.402823466e38f; }

    for (int r = 0; r < RPW; ++r) {
        const v4f* row = (const v4f*)(xw + (size_t)r * C_) + lane * 2;
        const v4f v0 = row[0];
        const v4f v1 = row[1];

        // per-channel running stats (lane owns channels lane*8 .. lane*8+7)
        cs[0] += v0.x; cs[1] += v0.y; cs[2] += v0.z; cs[3] += v0.w;
        cs[4] += v1.x; cs[5] += v1.y; cs[6] += v1.z; cs[7] += v1.w;
        cm[0] = fmaxf(cm[0], v0.x); cm[1] = fmaxf(cm[1], v0.y);
        cm[2] = fmaxf(cm[2], v0.z); cm[3] = fmaxf(cm[3], v0.w);
        cm[4] = fmaxf(cm[4], v1.x); cm[5] = fmaxf(cm[5], v1.y);
        cm[6] = fmaxf(cm[6], v1.z); cm[7] = fmaxf(cm[7], v1.w);

        // spatial (over C) reduction: 8-wide local, then shfl across wave32
        float rs = ((v0.x + v0.y) + (v0.z + v0.w))
                 + ((v1.x + v1.y) + (v1.z + v1.w));
        float rm = fmaxf(fmaxf(fmaxf(v0.x, v0.y), fmaxf(v0.z, v0.w)),
                         fmaxf(fmaxf(v1.x, v1.y), fmaxf(v1.z, v1.w)));
        #pragma unroll
        for (int off = 16; off > 0; off >>= 1) {
            rs += __shfl_down(rs, off, 32);
            rm  = fmaxf(rm, __shfl_down(rm, off, 32));
        }
        if (lane == 0) {
            __builtin_nontemporal_store(rs * (1.0f / C_),
                                        &avg_s[(size_t)b * L_ + l0 + r]);
            __builtin_nontemporal_store(rm,
                                        &max_s[(size_t)b * L_ + l0 + r]);
        }
    }

    // combine per-wave channel partials once via LDS
    #pragma unroll
    for (int j = 0; j < 8; ++j) {
        sSum[wv][lane * 8 + j] = cs[j];
        sMax[wv][lane * 8 + j] = cm[j];
    }
    __syncthreads();

    const int c = threadIdx.x;
    float s = 0.0f;
    float m = -3.402823466e38f;
    #pragma unroll
    for (int w = 0; w < 8; ++w) {
        s += sSum[w][c];
        m  = fmaxf(m, sMax[w][c]);
    }
    __builtin_nontemporal_store(s, &psum[((size_t)b * NLB + lb) * C_ + c]);
    __builtin_nontemporal_store(m, &pmax[((size_t)b * NLB + lb) * C_ + c]);
}

// ---------------------------------------------------------------------------
// Kernel 2: fold channel partials -> stats matrix [64, C_]
//           rows 0..31  = mean over L  (per batch)
//           rows 32..63 = max  over L
// ---------------------------------------------------------------------------
__global__ __launch_bounds__(256) void finalize_kernel(
    const float* __restrict__ psum,
    const float* __restrict__ pmax,
    float* __restrict__ stats)          // [64, C_]
{
    const int tid = blockIdx.x * 256 + threadIdx.x;   // 0 .. B_*C_-1
    const int b = tid / C_;
    const int c = tid % C_;
    float s = 0.0f;
    float m = -3.402823466e38f;
    for (int lb = 0; lb < NLB; ++lb) {
        s += psum[((size_t)b * NLB + lb) * C_ + c];
        m  = fmaxf(m, pmax[((size_t)b * NLB + lb) * C_ + c]);
    }
    stats[(size_t)b * C_ + c]        = s * (1.0f / L_);
    stats[(size_t)(b + 32) * C_ + c] = m;
}

// ---------------------------------------------------------------------------
// Kernel 3: 7-tap, 2-input-channel SAME conv over spatial stats + sigmoid.
// conv_w layout (W, I, O) = [7,2,1] -> w[k*2 + i]
// ---------------------------------------------------------------------------
__global__ __launch_bounds__(256) void conv_kernel(
    const float* __restrict__ avg_s,
    const float* __restrict__ max_s,
    const float* __restrict__ cw,       // [14]
    float* __restrict__ sig_s)          // [B_, L_]
{
    const int tid = blockIdx.x * 256 + threadIdx.x;   // 0 .. B_*L_-1
    const int b = tid / L_;
    const int l = tid % L_;
    float acc = 0.0f;
    #pragma unroll
    for (int k = 0; k < 7; ++k) {
        const int ll = l + k - 3;
        if (ll >= 0 && ll < L_) {
            acc += avg_s[(size_t)b * L_ + ll] * cw[k * 2 + 0]
                 + max_s[(size_t)b * L_ + ll] * cw[k * 2 + 1];
        }
    }
    sig_s[tid] = 1.0f / (1.0f + __expf(-acc));
}

// ---------------------------------------------------------------------------
// Kernel 4: channel-attention MLP on the tensor core path.
// One wave32 workgroup.  D = A x B + C with V_WMMA_F32_16X16X4_F32.
//   GEMM1: [64,256] stats @ [256,16] W1  (two 16-row tiles per branch,
//          separate accumulators for avg/max since ReLU is applied per branch)
//   branch-sum: (relu(Ha)+relu(Hm)) -> Hsum[32,16] in LDS
//          (valid since (Ha@W2)+(Hm@W2) == (Ha+Hm)@W2)
//   GEMM2: [32,16] Hsum @ [16,256] W2, + 2*b2, sigmoid -> att_c[32,256]
// Fragment layouts per ISA 7.12.2:
//   A 16x4 f32 : lanes 0-15 -> {K=0,K=1}, lanes 16-31 -> {K=2,K=3}
//   B 4x16 f32 : row striped across lanes (same K split as A)
//   C/D 16x16  : VGPR j -> row j (lanes 0-15) / row 8+j (lanes 16-31)
// ---------------------------------------------------------------------------
__global__ __launch_bounds__(32) void mlp_wmma_kernel(
    const float* __restrict__ stats,    // [64, C_]
    const float* __restrict__ W1,       // [C_, H_]
    const float* __restrict__ b1,       // [H_]
    const float* __restrict__ W2,       // [H_, C_]
    const float* __restrict__ b2,       // [C_]
    float* __restrict__ att_c)          // [B_, C_]
{
    __shared__ float Hsum[32][H_];

    const int  lane = threadIdx.x;      // 0..31
    const bool lo   = lane < 16;
    const int  half = lane & 15;

    // ---------------- GEMM1 + ReLU + branch sum ----------------
    for (int mt = 0; mt < 2; ++mt) {
        v8f accA = {};
        v8f accM = {};
        const int rowA = mt * 16 + half;      // avg branch row (0..31)
        const int rowM = rowA + 32;           // max branch row (32..63)
        for (int k0 = 0; k0 < C_; k0 += 4) {
            const int ka = lo ? k0 : (k0 + 2);
            v2f a, am, bf;
            a.x  = stats[(size_t)rowA * C_ + ka];
            a.y  = stats[(size_t)rowA * C_ + ka + 1];
            am.x = stats[(size_t)rowM * C_ + ka];
            am.y = stats[(size_t)rowM * C_ + ka + 1];
            bf.x = W1[(size_t)ka * H_ + half];
            bf.y = W1[(size_t)(ka + 1) * H_ + half];
            accA = __builtin_amdgcn_wmma_f32_16x16x4_f32(
                       false, a,  false, bf, (short)0, accA, false, false);
            accM = __builtin_amdgcn_wmma_f32_16x16x4_f32(
                       false, am, false, bf, (short)0, accM, false, false);
        }
        #pragma unroll
        for (int j = 0; j < 8; ++j) {
            const int m = mt * 16 + (lo ? j : (8 + j));   // batch index 0..31
            const int n = half;                            // hidden index
            const float hA = fmaxf(accA[j] + b1[n], 0.0f);
            const float hM = fmaxf(accM[j] + b1[n], 0.0f);
            Hsum[m][n] = hA + hM;
        }
    }
    __syncthreads();

    // ---------------- GEMM2 + bias + sigmoid ----------------
    for (int mt = 0; mt < 2; ++mt) {
        const int rowH = mt * 16 + half;
        for (int nt = 0; nt < 16; ++nt) {
            v8f acc = {};
            #pragma unroll
            for (int k0 = 0; k0 < H_; k0 += 4) {
                const int ka = lo ? k0 : (k0 + 2);
                v2f a, bf;
                a.x = Hsum[rowH][ka];
                a.y = Hsum[rowH][ka + 1];
                const int n = nt * 16 + half;
                bf.x = W2[(size_t)ka * C_ + n];
                bf.y = W2[(size_t)(ka + 1) * C_ + n];
                acc = __builtin_amdgcn_wmma_f32_16x16x4_f32(
                          false, a, false, bf, (short)0, acc, false, false);
            }
            #pragma unroll
            for (int j = 0; j < 8; ++j) {
                const int m = mt * 16 + (lo ? j : (8 + j));  // batch
                const int c = nt * 16 + half;                // channel
                const float v = acc[j] + 2.0f * b2[c];
                att_c[(size_t)m * C_ + c] = 1.0f / (1.0f + __expf(-v));
            }
        }
    }
}

// ---------------------------------------------------------------------------
// Kernel 5: streaming combine.  out = x*x*(att_c[b,c] + sig_s[b,l])
// x read is its last use -> non-temporal load (hits L2 filled by kernel 1,
// no need to retain).  out store is non-temporal so the 128MB of stores do
// not evict x from L2 mid-pass.
// ---------------------------------------------------------------------------
__global__ __launch_bounds__(256) void combine_kernel(
    const float* __restrict__ x,
    const float* __restrict__ att_c,    // [B_, C_]
    const float* __restrict__ sig_s,    // [B_, L_]
    float* __restrict__ out)
{
    const size_t i  = (size_t)blockIdx.x * 256 + threadIdx.x;  // over B*L*C/4
    const size_t c4 = i & 63;        // which float4 within the channel row
    const size_t bl = i >> 6;        // b*L + l
    const size_t b  = bl >> 12;      // L_ = 4096

    const v4f* x4 = (const v4f*)x;
    const v4f* a4 = (const v4f*)att_c;
    v4f*       o4 = (v4f*)out;

    const v4f xv = __builtin_nontemporal_load(&x4[i]);
    const v4f av = a4[b * (C_ / 4) + c4];
    const float s = sig_s[bl];

    v4f r;
    r.x = xv.x * xv.x * (av.x + s);
    r.y = xv.y * xv.y * (av.y + s);
    r.z = xv.z * xv.z * (av.z + s);
    r.w = xv.w * xv.w * (av.w + s);
    __builtin_nontemporal_store(r, &o4[i]);
}

// ---------------------------------------------------------------------------
extern "C" void kernel_launch(void* const* d_in, const int* in_sizes, int n_in,
                              void* d_out, int out_size, void* d_ws, size_t ws_size,
                              hipStream_t stream) {
    const float* x      = (const float*)d_in[0];   // [B,L,C]
    const float* W1     = (const float*)d_in[1];   // [C,16]
    const float* b1     = (const float*)d_in[2];   // [16]
    const float* W2     = (const float*)d_in[3];   // [16,C]
    const float* b2     = (const float*)d_in[4];   // [C]
    const float* conv_w = (const float*)d_in[5];   // [7,2,1]
    float* out = (float*)d_out;

    // Workspace layout (floats)
    float* ws    = (float*)d_ws;
    float* psum  = ws;                              // B*NLB*C = 131072
    float* pmax  = psum + (size_t)B_ * NLB * C_;    // 131072
    float* avg_s = pmax + (size_t)B_ * NLB * C_;    // B*L = 131072
    float* max_s = avg_s + (size_t)B_ * L_;         // 131072
    float* sig_s = max_s + (size_t)B_ * L_;         // 131072
    float* stats = sig_s + (size_t)B_ * L_;         // 64*C = 16384
    float* att_c = stats + (size_t)64 * C_;         // B*C  = 8192

    // 1) fused reductions over L (per channel) and C (per position)
    reduce_kernel<<<B_ * NLB, 256, 0, stream>>>(x, psum, pmax, avg_s, max_s);

    // 2) fold channel partials into [64, C] stats (avg rows, then max rows)
    finalize_kernel<<<(B_ * C_) / 256, 256, 0, stream>>>(psum, pmax, stats);

    // 3) spatial conv + sigmoid
    conv_kernel<<<(B_ * L_) / 256, 256, 0, stream>>>(avg_s, max_s, conv_w, sig_s);

    // 4) channel MLP on WMMA path (one wave32 workgroup)
    mlp_wmma_kernel<<<1, 32, 0, stream>>>(stats, W1, b1, W2, b2, att_c);

    // 5) streaming combine
    combine_kernel<<<((size_t)B_ * L_ * C_ / 4) / 256, 256, 0, stream>>>(
        x, att_c, sig_s, out);
}